// VisibilityRebuildModule_89799176225489
// MI455X (gfx1250) — compile-verified
//
#include <hip/hip_runtime.h>
#include <stdint.h>

#define BATCH   64
#define HW      65536      // 256*256
#define SELN    200
#define EPSF    1e-8f

typedef float    v2f  __attribute__((ext_vector_type(2)));
typedef float    v4f  __attribute__((ext_vector_type(4)));
typedef float    v8f  __attribute__((ext_vector_type(8)));
typedef uint32_t u32x4 __attribute__((ext_vector_type(4)));
typedef int      i32x4 __attribute__((ext_vector_type(4)));
typedef int      i32x8 __attribute__((ext_vector_type(8)));

__device__ __forceinline__ uint32_t hash32(uint32_t x){
  x ^= x >> 16; x *= 0x7feb352dU;
  x ^= x >> 15; x *= 0x846ca68bU;
  x ^= x >> 16; return x;
}

// ---------------- Stage A: deterministic rejection sampling ----------------
__global__ __launch_bounds__(256) void sample_kernel(const float* __restrict__ kpt,
                                                     const float* __restrict__ fmask,
                                                     int* __restrict__ d_idx,
                                                     int* __restrict__ d_has){
  __shared__ int s_found[256];
  const int b = blockIdx.x, i = threadIdx.x;
  int found = 0, pick = 0;
  if (i < SELN){
    const float* kb = kpt + (size_t)b*3*HW;
    for (int tr = 0; tr < 64; ++tr){
      uint32_t h = hash32((uint32_t)(b*2654435761u) ^ (uint32_t)(i*0x85EBCA77u)
                          ^ (uint32_t)(tr*0xC2B2AE3Du) ^ 123u);
      int p = (int)(h & (HW-1));
      float fm = fmask[p];
      float s  = kb[p] + kb[HW+p] + kb[2*HW+p];
      if (fm > 0.f && s != 0.f){ found = 1; pick = p; break; }
    }
    d_idx[b*SELN + i] = pick;
  }
  s_found[i] = found;
  __syncthreads();
  for (int s = 128; s > 0; s >>= 1){
    if (i < s) s_found[i] |= s_found[i+s];
    __syncthreads();
  }
  if (i == 0) d_has[b] = s_found[0];
}

// ---------------- Stage B: per-batch statistics (block per batch) ----------
// wave32 shuffle reduction + one 8-slot LDS stage: 2 barriers per reduction
// instead of ~10 (cuts the s_wait/s_barrier bloat seen in round 1).
__device__ __forceinline__ float blockReduceSum(float v, float* s8){
  #pragma unroll
  for (int o = 16; o > 0; o >>= 1) v += __shfl_xor(v, o, 32);
  const int wv = threadIdx.x >> 5, ln = threadIdx.x & 31;
  __syncthreads();                 // protect s8 reuse across consecutive calls
  if (ln == 0) s8[wv] = v;
  __syncthreads();
  float r = 0.f;
  #pragma unroll
  for (int i = 0; i < 8; ++i) r += s8[i];
  return r;
}

__global__ __launch_bounds__(256) void stats_kernel(const float* __restrict__ Offset,
                                                    const float* __restrict__ kpt,
                                                    const float* __restrict__ conf,
                                                    const float* __restrict__ meanp,
                                                    const int* __restrict__ d_idx,
                                                    const int* __restrict__ d_has,
                                                    float* __restrict__ d_stats){
  __shared__ float s8[8];
  const int b = blockIdx.x, i = threadIdx.x;
  const int has = d_has[b];
  const float g = has ? 1.f : 0.f;
  float src[3] = {0,0,0}, dst[3] = {0,0,0};
  float w = 0.f;
  if (i < SELN){
    const int p = d_idx[b*SELN + i];
    const float* ob = Offset + (size_t)b*3*HW;
    const float* kb = kpt    + (size_t)b*3*HW;
    #pragma unroll
    for (int c = 0; c < 3; ++c){
      src[c] = (ob[c*HW + p] + meanp[c*HW + p]) * g;   // kpt_src (gated)
      dst[c] = kb[c*HW + p] * g;                        // kpt_dst (gated)
    }
    w = has ? conf[(size_t)b*HW + p] : 1.f;
  }
  const float sumW = blockReduceSum(w, s8) + 0.5f*EPSF;
  float csrc[3], cdst[3], mS[3], mD[3];
  #pragma unroll
  for (int c = 0; c < 3; ++c){
    csrc[c] = blockReduceSum(w*src[c], s8) / sumW;
    cdst[c] = blockReduceSum(w*dst[c], s8) / sumW;
    mS[c]   = blockReduceSum(src[c], s8) * (1.f/(float)SELN);
    mD[c]   = blockReduceSum(dst[c], s8) * (1.f/(float)SELN);
  }
  float n1 = 0.f, n2 = 0.f;
  if (i < SELN){
    float dx0 = src[0]-csrc[0], dy0 = src[1]-csrc[1], dz0 = src[2]-csrc[2];
    float dx1 = dst[0]-cdst[0], dy1 = dst[1]-cdst[1], dz1 = dst[2]-cdst[2];
    n1 = sqrtf(dx0*dx0 + dy0*dy0 + dz0*dz0);
    n2 = sqrtf(dx1*dx1 + dy1*dy1 + dz1*dz1);
  }
  const float sd1 = blockReduceSum(n1, s8) + 0.5f*EPSF;
  const float sd2 = blockReduceSum(n2, s8) + 0.5f*EPSF;
  const float S = sd2 / sd1;
  float AA[3], BB[3];
  #pragma unroll
  for (int c = 0; c < 3; ++c){
    AA[c] = S*(src[c]-mS[c]);
    BB[c] = dst[c]-mD[c];
  }
  float Hm[9];
  #pragma unroll
  for (int c = 0; c < 3; ++c)
    #pragma unroll
    for (int d = 0; d < 3; ++d)
      Hm[c*3+d] = blockReduceSum(w*AA[c]*BB[d], s8);    // w==0 for i>=SELN
  if (i == 0){
    float* st = d_stats + b*16;
    #pragma unroll
    for (int k = 0; k < 9; ++k) st[k] = Hm[k];
    st[9]  = S;
    st[10] = mD[0]; st[11] = mD[1]; st[12] = mD[2];
    st[13] = S*mS[0]; st[14] = S*mS[1]; st[15] = S*mS[2];
  }
}

// ---------------- Stage C: 3x3 SVD (Jacobi on H^T H), R, t, packed B -------
__global__ void solve_kernel(const float* __restrict__ d_stats,
                             float* __restrict__ d_bmat,
                             float* __restrict__ outR){
  const int b = threadIdx.x;
  if (b >= BATCH) return;
  const float* st = d_stats + b*16;
  float Hm[3][3];
  for (int i = 0; i < 3; ++i)
    for (int j = 0; j < 3; ++j)
      Hm[i][j] = st[i*3+j];
  const float S = st[9];
  const float mD[3] = {st[10], st[11], st[12]};
  const float mA[3] = {st[13], st[14], st[15]};

  float G[3][3], V[3][3] = {{1,0,0},{0,1,0},{0,0,1}};
  for (int i = 0; i < 3; ++i)
    for (int j = 0; j < 3; ++j){
      float acc = 0.f;
      for (int k = 0; k < 3; ++k) acc += Hm[k][i]*Hm[k][j];
      G[i][j] = acc;
    }
  const int PP[3] = {0,0,1}, QQ[3] = {1,2,2};
  for (int sweep = 0; sweep < 12; ++sweep){
    for (int pi = 0; pi < 3; ++pi){
      const int p = PP[pi], q = QQ[pi];
      const float apq = G[p][q];
      if (fabsf(apq) < 1e-30f) continue;
      const float tau = (G[q][q]-G[p][p])/(2.f*apq);
      const float tt  = (tau >= 0.f ? 1.f : -1.f)/(fabsf(tau)+sqrtf(1.f+tau*tau));
      const float cc  = rsqrtf(1.f+tt*tt);
      const float ssn = tt*cc;
      for (int k = 0; k < 3; ++k){ float a=G[k][p], c2=G[k][q]; G[k][p]=cc*a-ssn*c2; G[k][q]=ssn*a+cc*c2; }
      for (int k = 0; k < 3; ++k){ float a=G[p][k], c2=G[q][k]; G[p][k]=cc*a-ssn*c2; G[q][k]=ssn*a+cc*c2; }
      for (int k = 0; k < 3; ++k){ float a=V[k][p], c2=V[k][q]; V[k][p]=cc*a-ssn*c2; V[k][q]=ssn*a+cc*c2; }
    }
  }
  float sv[3];
  for (int j = 0; j < 3; ++j) sv[j] = sqrtf(fmaxf(G[j][j], 0.f));
  float U[3][3]; int bad = -1;
  for (int j = 0; j < 3; ++j){
    float u[3];
    for (int k = 0; k < 3; ++k)
      u[k] = Hm[k][0]*V[0][j] + Hm[k][1]*V[1][j] + Hm[k][2]*V[2][j];
    if (sv[j] > 1e-12f){ for (int k = 0; k < 3; ++k) U[k][j] = u[k]/sv[j]; }
    else               { bad = j; for (int k = 0; k < 3; ++k) U[k][j] = 0.f; }
  }
  if (bad >= 0){
    const int j1 = (bad+1)%3, j2 = (bad+2)%3;
    const float cx = U[1][j1]*U[2][j2]-U[2][j1]*U[1][j2];
    const float cy = U[2][j1]*U[0][j2]-U[0][j1]*U[2][j2];
    const float cz = U[0][j1]*U[1][j2]-U[1][j1]*U[0][j2];
    const float len = sqrtf(cx*cx+cy*cy+cz*cz);
    if (len > 1e-20f){ U[0][bad]=cx/len; U[1][bad]=cy/len; U[2][bad]=cz/len; }
  }
  float R[3][3];
  for (int i = 0; i < 3; ++i)
    for (int k = 0; k < 3; ++k){
      float acc = 0.f;
      for (int j = 0; j < 3; ++j) acc += V[i][j]*U[k][j];
      R[i][k] = acc;
    }
  float tv[3];
  for (int k = 0; k < 3; ++k)
    tv[k] = mD[k] - (mA[0]*R[k][0] + mA[1]*R[k][1] + mA[2]*R[k][2]);
  for (int i = 0; i < 3; ++i)
    for (int k = 0; k < 3; ++k)
      outR[b*9 + i*3 + k] = R[i][k];

  // 4x16 WMMA B operand, packed into the per-lane ISA layout:
  // VGPR0: lanes0-15 K=0, lanes16-31 K=2; VGPR1: lanes0-15 K=1, lanes16-31 K=3.
  float Bv[4][16];
  for (int c = 0; c < 4; ++c)
    for (int nn = 0; nn < 16; ++nn) Bv[c][nn] = 0.f;
  for (int nn = 0; nn < 3; ++nn){
    for (int c = 0; c < 3; ++c){
      Bv[c][nn]   = S*R[nn][c];   // cols 0..2: outpos (t folded via K=3 row)
      Bv[c][nn+3] = S*R[nn][c];   // cols 3..5: shape_map (no t)
    }
    Bv[3][nn] = tv[nn];
  }
  float* pk = d_bmat + b*64;
  for (int lane = 0; lane < 32; ++lane){
    const int nn = lane & 15, hi = lane >> 4;
    pk[lane]    = Bv[hi ? 2 : 0][nn];
    pk[32+lane] = Bv[hi ? 3 : 1][nn];
  }
}

// ---------------- Stage D: bulk transform, 4x WMMA + TDM-resident B --------
// 64-pixel super-tile per wave iteration: 4 back-to-back WMMAs sharing one B
// operand (DMA'd into LDS once per block by the Tensor Data Mover), LDS
// transpose staging, then 3 coalesced global_store_b128 per lane.
__global__ __launch_bounds__(256) void transform_kernel(const float* __restrict__ Offset,
                                                        const float* __restrict__ meanp,
                                                        const float* __restrict__ d_bmat,
                                                        float* __restrict__ out){
  __shared__ float s_bmat[BATCH*64];     // 16 KB: all per-batch B operands
  __shared__ float s_stage[8*384];       // 12 KB: per-wave D staging
  const int lane = threadIdx.x & 31;
  const int wave = threadIdx.x >> 5;
  const int n  = lane & 15;
  const int hi = lane >> 4;
  const int c0 = hi ? 2 : 0;

  // ---- TDM: one 16 KB tensor DMA (wave 0 issues; EXEC ignored by TDM) ----
  if (threadIdx.x < 32){
    const uint32_t ldsAddr = (uint32_t)(uintptr_t)(&s_bmat[0]);
    const uint64_t ga = (uint64_t)(uintptr_t)d_bmat;
    u32x4 g0;
    g0[0] = 1u;                                   // count=1 (valid), no gather
    g0[1] = ldsAddr;                              // lds_addr
    g0[2] = (uint32_t)(ga & 0xFFFFFFFFu);         // global_addr[31:0]
    g0[3] = (uint32_t)((ga >> 32) & 0x01FFFFFFu)  // global_addr[56:32]
            | 0x80000000u;                        // type=2 ("image")
    i32x8 g1;
    g1[0] = 0x00020000;                           // data_size=4B, mask=0
    g1[1] = (int)((BATCH*64) << 16);              // tensor_dim0[15:0]=4096 @63:48
    g1[2] = (int)(1 << 16);                       // tensor_dim1=1 @111:80
    g1[3] = (int)((BATCH*64) << 16);              // tile_dim0=4096 @127:112
    g1[4] = 1;                                    // tile_dim1=1 @143:128
    g1[5] = BATCH*64;                             // tensor_dim0_stride=4096
    g1[6] = 0;
    g1[7] = 0;
    i32x4 gz = {0,0,0,0};
#if defined(__clang_major__) && (__clang_major__ >= 23)
    i32x8 gz8 = {0,0,0,0,0,0,0,0};
    __builtin_amdgcn_tensor_load_to_lds(g0, g1, gz, gz, gz8, 0);
#else
    __builtin_amdgcn_tensor_load_to_lds(g0, g1, gz, gz, 0);
#endif
    __builtin_amdgcn_s_wait_tensorcnt(0);
  }
  __syncthreads();

  const int nSuper = BATCH * (HW/64);     // 65536 super-tiles
  const int nSlots = gridDim.x * 8;       // 16384 -> 4 uniform iterations
  float* myLds = s_stage + wave*384;
  const size_t SHAPE_OFF = (size_t)BATCH*3*HW - (size_t)3*HW;

  for (int sp = blockIdx.x*8 + wave; sp < nSuper; sp += nSlots){
    const int b     = sp >> 10;           // 1024 super-tiles per batch image
    const int pbase = (sp & 1023) << 6;
    const float* ob = Offset + (size_t)b*3*HW;

    const int sp2 = sp + nSlots;          // prefetch next iteration's A data
    if (sp2 < nSuper){
      const int b2 = sp2 >> 10, p2 = ((sp2 & 1023) << 6) + n;
      __builtin_prefetch(Offset + (size_t)b2*3*HW + (size_t)c0*HW + p2, 0, 0);
    }

    // B operand: two coalesced LDS reads from the TDM-resident table
    v2f Bm; Bm.x = s_bmat[b*64 + lane]; Bm.y = s_bmat[b*64 + 32 + lane];

    v8f D[4];
    #pragma unroll
    for (int s = 0; s < 4; ++s){
      const int p = pbase + s*16 + n;
      float a0 = ob[(size_t)c0*HW + p] + meanp[(size_t)c0*HW + p];
      float a1 = ob[(size_t)HW   + p] + meanp[(size_t)HW   + p];
      if (hi) a1 = 1.0f;                  // K=3 row of A = ones column
      v2f A; A.x = a0; A.y = a1;
      v8f C = {};
      D[s] = __builtin_amdgcn_wmma_f32_16x16x4_f32(
          false, A, false, Bm, (short)0, C, false, false);
    }

    // Stage: lds[ch*64 + pixel] so each channel is a 64-float contiguous run
    if (n < 6){
      #pragma unroll
      for (int s = 0; s < 4; ++s){
        const int base = n*64 + s*16 + 8*hi;   // M = v + 8*hi
        #pragma unroll
        for (int v = 0; v < 8; ++v) myLds[base + v] = D[s][v];
      }
    }
    __syncthreads();

    // 3x b128 per lane: half-wave writes 256B contiguous per channel plane.
    // Branchless outpos/shape region select (regions adjacent in d_out).
    #pragma unroll
    for (int j = 0; j < 3; ++j){
      const int idx = j*128 + lane*4;
      const int ch  = idx >> 6;           // = 2*j + hi
      const int rem = idx & 63;
      v4f val = *(const v4f*)(myLds + idx);
      size_t o = (size_t)b*3*HW + (size_t)ch*HW + (size_t)(pbase + rem);
      o += SHAPE_OFF * (size_t)(ch >= 3);
      *(v4f*)(out + o) = val;
    }
    __syncthreads();
  }
}

// ---------------------------------------------------------------------------
extern "C" void kernel_launch(void* const* d_in, const int* in_sizes, int n_in,
                              void* d_out, int out_size, void* d_ws, size_t ws_size,
                              hipStream_t stream) {
  (void)in_sizes; (void)n_in; (void)out_size; (void)ws_size;
  const float* Offset = (const float*)d_in[0];
  const float* Posmap = (const float*)d_in[1];
  const float* conf   = (const float*)d_in[2];
  const float* meanp  = (const float*)d_in[3];
  const float* fmask  = (const float*)d_in[4];

  float* out  = (float*)d_out;
  float* outR = out + 2 * (size_t)BATCH*3*HW;

  char* ws = (char*)d_ws;                    // ~92 KB total
  int*   d_idx   = (int*)  ws;               // BATCH*SELN ints
  int*   d_has   = (int*)  (ws + 64*1024);   // BATCH ints
  float* d_stats = (float*)(ws + 68*1024);   // BATCH*16 floats
  float* d_bmat  = (float*)(ws + 76*1024);   // BATCH*64 floats (TDM source)

  sample_kernel   <<<BATCH, 256, 0, stream>>>(Posmap, fmask, d_idx, d_has);
  stats_kernel    <<<BATCH, 256, 0, stream>>>(Offset, Posmap, conf, meanp, d_idx, d_has, d_stats);
  solve_kernel    <<<1,     64,  0, stream>>>(d_stats, d_bmat, outR);
  transform_kernel<<<2048,  256, 0, stream>>>(Offset, meanp, d_bmat, out);
}